// MotionE_85117661872152
// MI455X (gfx1250) — compile-verified
//
#include <hip/hip_runtime.h>
#include <hip/hip_bf16.h>

// Problem constants (from reference)
#define L_SEQ   512
#define BATCH   32
#define DMODEL  256
#define NSTATE  16
#define DTRANK  16
#define DINNER  256
#define NROWS   (L_SEQ * BATCH)   // 16384 rows, row index r = l*BATCH + b
#define XDBL_W  48                // dt_rank + 2*n_state
#define CHUNK   32                // scan chunk length
#define NCHUNK  (L_SEQ / CHUNK)   // 16 chunks

typedef float v2f __attribute__((ext_vector_type(2)));
typedef float v8f __attribute__((ext_vector_type(8)));

// Async memory->LDS B128 copy (gfx1250): LDS written directly, tracked by
// ASYNCcnt (CDNA5 ISA).  ldsoff is the LDS byte offset
// (flat-pointer low 32 bits, per aperture rule), gaddr the global address.
__device__ __forceinline__ void async_ld_lds_b128(unsigned ldsoff,
                                                  unsigned long long gaddr) {
  asm volatile("global_load_async_to_lds_b128 %0, %1, off"
               :: "v"(ldsoff), "v"(gaddr) : "memory");
}
__device__ __forceinline__ void wait_asynccnt0() {
  asm volatile("s_wait_asynccnt 0x0" ::: "memory");
}

// ---------------------------------------------------------------------------
// Wave-level fp32 WMMA: NT adjacent 16x16 tiles sharing one A fragment.
//   C[t] = A[r0:r0+16, :K] * W[n0+16t : n0+16t+16, :K]^T
// V_WMMA_F32_16X16X4_F32, K stepped by 4.  Per k-step: 2 A floats + 2*NT B
// floats feed NT WMMAs (2.5 floats/WMMA at NT=4).
// A layout (ISA 7.12.2): lane<16 -> M=lane, K={k0,k0+1}; lane>=16 -> K={k0+2,k0+3}.
// C/D layout: VGPR i -> M = i + 8*(lane>>4), N = lane&15.
// ---------------------------------------------------------------------------
template <int NT>
__device__ __forceinline__ void wmma_tiles_f32(const float* __restrict__ A, int lda,
                                               const float* __restrict__ W, int ldw,
                                               int r0, int n0, int K, v8f* acc) {
  const int lane = (int)(threadIdx.x & 31u);
  const int half = lane >> 4;
  const int lr   = lane & 15;
  const float* __restrict__ arow = A + (size_t)(r0 + lr) * (size_t)lda;
  const float* wrow[NT];
#pragma unroll
  for (int t = 0; t < NT; ++t)
    wrow[t] = W + (size_t)(n0 + 16 * t + lr) * (size_t)ldw;
  const v8f zero = {0.f, 0.f, 0.f, 0.f, 0.f, 0.f, 0.f, 0.f};
#pragma unroll
  for (int t = 0; t < NT; ++t) acc[t] = zero;
#pragma unroll 2
  for (int k0 = 0; k0 < K; k0 += 4) {
    const int ka = k0 + 2 * half;
    v2f a; a.x = arow[ka]; a.y = arow[ka + 1];
#pragma unroll
    for (int t = 0; t < NT; ++t) {
      v2f b; b.x = wrow[t][ka]; b.y = wrow[t][ka + 1];
      // 8 args: (neg_a, A, neg_b, B, c_mod, C, reuse_a, reuse_b)
      acc[t] = __builtin_amdgcn_wmma_f32_16x16x4_f32(false, a, false, b,
                                                     (short)0, acc[t],
                                                     false, false);
    }
  }
}

// ---------------------------------------------------------------------------
// Kernel 1: x_dbl = flow @ x_proj_w^T   (16384 x 256)*(48 x 256)^T
// One wave covers the full 16x48 row-tile (NT=3).
// ---------------------------------------------------------------------------
__global__ void xproj_kernel(const float* __restrict__ flow,
                             const float* __restrict__ xw,
                             float* __restrict__ xdbl) {
  const int rt = (int)blockIdx.x * 8 + (int)(threadIdx.x >> 5);  // row tile 0..1023
  const int r0 = rt * 16;
  v8f acc[3];
  wmma_tiles_f32<3>(flow, DINNER, xw, DINNER, r0, 0, DINNER, acc);
  const int lane = (int)(threadIdx.x & 31u);
  const int half = lane >> 4, lr = lane & 15;
#pragma unroll
  for (int t = 0; t < 3; ++t)
#pragma unroll
    for (int i = 0; i < 8; ++i) {
      const int m = i + 8 * half;
      xdbl[(size_t)(r0 + m) * XDBL_W + 16 * t + lr] = acc[t][i];
    }
}

// ---------------------------------------------------------------------------
// Kernel 2: delta = softplus(x_dbl[:, :16] @ dt_proj_w^T + dt_proj_b)
// (16384 x 16)*(256 x 16)^T; K=16 -> 4 WMMA steps; wave covers 16x64 (NT=4).
// ---------------------------------------------------------------------------
__global__ void delta_kernel(const float* __restrict__ xdbl,
                             const float* __restrict__ dtw,
                             const float* __restrict__ dtb,
                             float* __restrict__ delta) {
  const int tile = (int)blockIdx.x * 8 + (int)(threadIdx.x >> 5);  // 0..4095
  const int r0 = (tile >> 2) * 16;
  const int n0 = (tile & 3) * 64;
  v8f acc[4];
  wmma_tiles_f32<4>(xdbl, XDBL_W, dtw, DTRANK, r0, n0, DTRANK, acc);
  const int lane = (int)(threadIdx.x & 31u);
  const int half = lane >> 4, lr = lane & 15;
#pragma unroll
  for (int t = 0; t < 4; ++t) {
    const int col = n0 + 16 * t + lr;
    const float bias = dtb[col];
#pragma unroll
    for (int i = 0; i < 8; ++i) {
      const int m = i + 8 * half;
      const float x = acc[t][i] + bias;
      const float sp = (x > 20.f) ? x : __logf(1.f + __expf(x));  // softplus
      delta[(size_t)(r0 + m) * DINNER + col] = sp;
    }
  }
}

// ---------------------------------------------------------------------------
// Scan pass A: per-chunk local scan from h=0; emit chunk-end state and sum(dt).
// Chunk operator product of dA collapses: prod_l exp(dt_l*A_n) = exp(A_n*S).
// Grid (BATCH, NCHUNK), block DINNER.  Chunk B-vectors (2 KB) staged via
// async memory->LDS B128 copies (no VGPR bounce), overlapping the exp preamble.
// ---------------------------------------------------------------------------
__global__ void scan_chunk_local(const float* __restrict__ u,      // (L,B,D)
                                 const float* __restrict__ xdbl,   // (r,48)
                                 const float* __restrict__ delta,  // (r,256)
                                 const float* __restrict__ A_log,  // (D,N)
                                 float* __restrict__ hloc,         // (NC,B,D,N)
                                 float* __restrict__ dsum) {       // (NC,B,D)
  const int b = (int)blockIdx.x;
  const int c = (int)blockIdx.y;
  const int d = (int)threadIdx.x;
  __shared__ __align__(16) float sB[CHUNK * NSTATE];   // 32 steps x 16 B
  // 2 KB = 128 B128 transfers; threads 0..127 issue one each (per-lane EXEC ok)
  if ((int)threadIdx.x < CHUNK * NSTATE / 4) {
    const int t = (int)threadIdx.x >> 2, seg = (int)threadIdx.x & 3;
    const size_t r = (size_t)(c * CHUNK + t) * BATCH + b;
    async_ld_lds_b128(
        (unsigned)(uintptr_t)&sB[t * NSTATE + seg * 4],
        (unsigned long long)(uintptr_t)(xdbl + r * XDBL_W + DTRANK + seg * 4));
  }
  float An[NSTATE], h[NSTATE];
#pragma unroll
  for (int n = 0; n < NSTATE; ++n) {
    An[n] = -__expf(A_log[d * NSTATE + n]);
    h[n] = 0.f;
  }
  wait_asynccnt0();
  __syncthreads();
  float S = 0.f;
  for (int t = 0; t < CHUNK; ++t) {
    const size_t r = (size_t)(c * CHUNK + t) * BATCH + b;
    const float dt = delta[r * DINNER + d];
    const float uv = u[r * DMODEL + d];
    const float dtu = dt * uv;
    S += dt;
#pragma unroll
    for (int n = 0; n < NSTATE; ++n)
      h[n] = __expf(dt * An[n]) * h[n] + dtu * sB[t * NSTATE + n];
  }
  const size_t base = ((size_t)c * BATCH + b) * DINNER + d;
#pragma unroll
  for (int n = 0; n < NSTATE; ++n) hloc[base * NSTATE + n] = h[n];
  dsum[base] = S;
}

// ---------------------------------------------------------------------------
// Scan pass B: sequential chunk-operator composition (tiny: 16 chunks).
// Emits exact chunk entry states h_in and the final state h_final (output 2).
// ---------------------------------------------------------------------------
__global__ void scan_combine(const float* __restrict__ h0,      // (B,D,N)
                             const float* __restrict__ A_log,   // (D,N)
                             const float* __restrict__ hloc,    // (NC,B,D,N)
                             const float* __restrict__ dsum,    // (NC,B,D)
                             float* __restrict__ hin,           // (NC,B,D,N)
                             float* __restrict__ hout) {        // (B,D,N)
  const int b = (int)blockIdx.x;
  const int d = (int)threadIdx.x;
  float An[NSTATE], h[NSTATE];
#pragma unroll
  for (int n = 0; n < NSTATE; ++n) {
    An[n] = -__expf(A_log[d * NSTATE + n]);
    h[n] = h0[((size_t)b * DINNER + d) * NSTATE + n];
  }
  for (int c = 0; c < NCHUNK; ++c) {
    const size_t base = ((size_t)c * BATCH + b) * DINNER + d;
#pragma unroll
    for (int n = 0; n < NSTATE; ++n) hin[base * NSTATE + n] = h[n];
    const float S = dsum[base];
#pragma unroll
    for (int n = 0; n < NSTATE; ++n)
      h[n] = __expf(An[n] * S) * h[n] + hloc[base * NSTATE + n];
  }
#pragma unroll
  for (int n = 0; n < NSTATE; ++n)
    hout[((size_t)b * DINNER + d) * NSTATE + n] = h[n];
}

// ---------------------------------------------------------------------------
// Scan pass C: re-scan each chunk from exact h_in, emitting y (+ skip).
// Chunk B+C vectors (4 KB) staged via async B128: exactly one transfer per
// thread (no divergence), then s_wait_asynccnt + one barrier per chunk.
// ---------------------------------------------------------------------------
__global__ void scan_chunk_emit(const float* __restrict__ u,
                                const float* __restrict__ xdbl,
                                const float* __restrict__ delta,
                                const float* __restrict__ A_log,
                                const float* __restrict__ Dp,
                                const float* __restrict__ hin,   // (NC,B,D,N)
                                float* __restrict__ Y) {         // (r,256)
  const int b = (int)blockIdx.x;
  const int c = (int)blockIdx.y;
  const int d = (int)threadIdx.x;
  __shared__ __align__(16) float sBC[CHUNK * 2 * NSTATE];  // per step: 16 B, 16 C
  {
    // 4 KB = 256 B128 transfers, one per thread
    const int t = (int)threadIdx.x >> 3, seg = (int)threadIdx.x & 7;
    const size_t r = (size_t)(c * CHUNK + t) * BATCH + b;
    async_ld_lds_b128(
        (unsigned)(uintptr_t)&sBC[t * 32 + seg * 4],
        (unsigned long long)(uintptr_t)(xdbl + r * XDBL_W + DTRANK + seg * 4));
  }
  float An[NSTATE], h[NSTATE];
  const size_t base = ((size_t)c * BATCH + b) * DINNER + d;
#pragma unroll
  for (int n = 0; n < NSTATE; ++n) {
    An[n] = -__expf(A_log[d * NSTATE + n]);
    h[n] = hin[base * NSTATE + n];
  }
  const float dskip = Dp[d];
  wait_asynccnt0();
  __syncthreads();
  for (int t = 0; t < CHUNK; ++t) {
    const size_t r = (size_t)(c * CHUNK + t) * BATCH + b;
    const float dt = delta[r * DINNER + d];
    const float uv = u[r * DMODEL + d];
    const float dtu = dt * uv;
    float y = 0.f;
#pragma unroll
    for (int n = 0; n < NSTATE; ++n) {
      h[n] = __expf(dt * An[n]) * h[n] + dtu * sBC[t * 32 + n];
      y += h[n] * sBC[t * 32 + NSTATE + n];
    }
    Y[r * DINNER + d] = y + uv * dskip;
  }
}

// ---------------------------------------------------------------------------
// Kernel 4: out = Y @ out_proj_w^T, scattered to (b, l, m) order (flatten(1)).
// Dominant GEMM (2.1 GF); wave covers 16x64 (NT=4).
// ---------------------------------------------------------------------------
__global__ void outproj_kernel(const float* __restrict__ Y,
                               const float* __restrict__ ow,
                               float* __restrict__ out) {
  const int tile = (int)blockIdx.x * 8 + (int)(threadIdx.x >> 5);  // 0..4095
  const int r0 = (tile >> 2) * 16;
  const int n0 = (tile & 3) * 64;
  v8f acc[4];
  wmma_tiles_f32<4>(Y, DINNER, ow, DINNER, r0, n0, DINNER, acc);
  const int lane = (int)(threadIdx.x & 31u);
  const int half = lane >> 4, lr = lane & 15;
#pragma unroll
  for (int t = 0; t < 4; ++t)
#pragma unroll
    for (int i = 0; i < 8; ++i) {
      const int m = i + 8 * half;
      const int r = r0 + m;           // r = l*BATCH + b
      const int l = r / BATCH;
      const int b = r - l * BATCH;
      out[((size_t)b * L_SEQ + l) * DMODEL + n0 + 16 * t + lr] = acc[t][i];
    }
}

// ---------------------------------------------------------------------------
extern "C" void kernel_launch(void* const* d_in, const int* in_sizes, int n_in,
                              void* d_out, int out_size, void* d_ws, size_t ws_size,
                              hipStream_t stream) {
  const float* pos   = (const float*)d_in[0];  // position_features (L,B,D)
  const float* flow  = (const float*)d_in[1];  // flow_features     (L,B,D)
  const float* h0    = (const float*)d_in[2];  // h_state           (B,D,N)
  const float* xw    = (const float*)d_in[3];  // x_proj_w          (48,256)
  const float* dtw   = (const float*)d_in[4];  // dt_proj_w         (256,16)
  const float* dtb   = (const float*)d_in[5];  // dt_proj_b         (256,)
  const float* A_log = (const float*)d_in[6];  // A_log             (256,16)
  const float* Dp    = (const float*)d_in[7];  // D_param           (256,)
  const float* ow    = (const float*)d_in[8];  // out_proj_w        (256,256)

  float* out  = (float*)d_out;                              // (B, L*D)
  float* hout = out + (size_t)BATCH * L_SEQ * DMODEL;       // (B, D, N)

  float* xdbl  = (float*)d_ws;                              // 16384*48
  float* delta = xdbl + (size_t)NROWS * XDBL_W;             // 16384*256
  float* Yw    = delta + (size_t)NROWS * DINNER;            // 16384*256
  float* hloc  = Yw + (size_t)NROWS * DINNER;               // NC*B*D*N
  float* hin   = hloc + (size_t)NCHUNK * BATCH * DINNER * NSTATE;
  float* dsum  = hin + (size_t)NCHUNK * BATCH * DINNER * NSTATE;  // NC*B*D

  // 1) x_proj GEMM: 1024 row-tiles, each wave does full 16x48
  xproj_kernel<<<1024 / 8, 256, 0, stream>>>(flow, xw, xdbl);
  // 2) dt_proj GEMM + softplus: 1024 row-tiles x 4 col-groups
  delta_kernel<<<4096 / 8, 256, 0, stream>>>(xdbl, dtw, dtb, delta);
  // 3) chunked parallel scan: A (local), B (combine), C (emit)
  scan_chunk_local<<<dim3(BATCH, NCHUNK), DINNER, 0, stream>>>(
      pos, xdbl, delta, A_log, hloc, dsum);
  scan_combine<<<BATCH, DINNER, 0, stream>>>(h0, A_log, hloc, dsum, hin, hout);
  scan_chunk_emit<<<dim3(BATCH, NCHUNK), DINNER, 0, stream>>>(
      pos, xdbl, delta, A_log, Dp, hin, Yw);
  // 4) out_proj GEMM (dominant): 1024 row-tiles x 4 col-groups
  outproj_kernel<<<4096 / 8, 256, 0, stream>>>(Yw, ow, out);
}